// SoftAlignModule_53017076302568
// MI455X (gfx1250) — compile-verified
//
#include <hip/hip_runtime.h>
#include <math.h>

#ifndef ENABLE_ASYNC_COPY
#define ENABLE_ASYNC_COPY 1
#endif

namespace {

constexpr int kC   = 256;
constexpr int kH   = 128;
constexpr int kW   = 128;
constexpr int kHW  = kH * kW;
constexpr int kN   = 4096;
constexpr int kA   = 5;
constexpr int kB   = 4;
constexpr int kT   = 16;      // point tile = WMMA M/N
constexpr int kLds = kC + 4;  // padded LDS row stride -> conflict-free WMMA frags

typedef float v2f __attribute__((ext_vector_type(2)));
typedef float v8f __attribute__((ext_vector_type(8)));
typedef int   v4i __attribute__((ext_vector_type(4)));

// AS-qualified pointer types for the async global->LDS builtin
// (param 1 is v4i in the global/__device__ AS per the compiler diagnostic).
typedef __attribute__((address_space(1))) v4i* gv4i_p;
typedef __attribute__((address_space(3))) v4i* lv4i_p;

#if defined(__HIP_DEVICE_COMPILE__) && __has_builtin(__builtin_amdgcn_wmma_f32_16x16x4_f32)
#define HAVE_WMMA_F32 1
#else
#define HAVE_WMMA_F32 0
#endif

#if ENABLE_ASYNC_COPY && defined(__HIP_DEVICE_COMPILE__) && \
    __has_builtin(__builtin_amdgcn_global_load_async_to_lds_b128)
#define USE_ASYNC_LDS 1
#else
#define USE_ASYNC_LDS 0
#endif

__device__ __forceinline__ float4 ld4(const float* p) { return *(const float4*)p; }

__device__ __forceinline__ void fma4(float4& a, float s, const float4& v) {
  a.x = fmaf(s, v.x, a.x); a.y = fmaf(s, v.y, a.y);
  a.z = fmaf(s, v.z, a.z); a.w = fmaf(s, v.w, a.w);
}

__device__ __forceinline__ int clampi(int v, int lo, int hi) {
  return v < lo ? lo : (v > hi ? hi : v);
}

// Fetch channels [c, c+4) of pixel `pix`. NHWC: one coalesced float4.
// NCHW fallback: 4 strided dword loads.
template<bool NHWC>
__device__ __forceinline__ float4 ldpix(const float* img, int pix, int c) {
  if (NHWC) {
    return ld4(img + (size_t)pix * kC + c);
  } else {
    const float* p = img + (size_t)c * kHW + pix;
    float4 r; r.x = p[0]; r.y = p[kHW]; r.z = p[2 * kHW]; r.w = p[3 * kHW];
    return r;
  }
}

#if USE_ASYNC_LDS
__device__ __forceinline__ void wait_async0() {
#if __has_builtin(__builtin_amdgcn_s_wait_asynccnt)
  __builtin_amdgcn_s_wait_asynccnt(0);
#else
  asm volatile("s_wait_asynccnt 0" ::: "memory");
#endif
}
#endif

// ---------------------------------------------------------------------------
// NCHW -> NHWC staging transpose ([B,C,HW] -> [B,HW,C]), 32x32 LDS tiles.
// ---------------------------------------------------------------------------
__global__ __launch_bounds__(256)
void transpose_nchw_nhwc(const float* __restrict__ src, float* __restrict__ dst) {
  __shared__ float tile[32][33];
  const int b   = blockIdx.z;
  const int hw0 = blockIdx.x * 32;
  const int c0  = blockIdx.y * 32;
  const float* s = src + (size_t)b * kC * kHW;
  float*       d = dst + (size_t)b * kHW * kC;
  for (int i = threadIdx.y; i < 32; i += 8)
    tile[i][threadIdx.x] = s[(size_t)(c0 + i) * kHW + hw0 + threadIdx.x];
  __syncthreads();
  for (int i = threadIdx.y; i < 32; i += 8)
    d[(size_t)(hw0 + i) * kC + c0 + threadIdx.x] = tile[threadIdx.x][i];
}

// ---------------------------------------------------------------------------
// Main fused kernel: one wave (32 lanes) handles a 16-point tile of one batch.
// ---------------------------------------------------------------------------
template<bool NHWC>
__global__ __launch_bounds__(32)
void soft_align_kernel(const float* __restrict__ imgF,   // x1 image
                       const float* __restrict__ imgE,   // pos_emb image
                       const float* __restrict__ pf,     // [B,N,C]
                       const float* __restrict__ pts,    // [B,N,2]
                       const float* __restrict__ ao,     // [B,A,N,2]
                       float* __restrict__ out)          // [B,N,C]
{
  __shared__ float s_rep [kT][kLds];
  __shared__ float s_feat[kT][kLds];

  const int lane = threadIdx.x;     // 0..31
  const int p    = lane & 15;       // my point in the tile
  const int half = lane >> 4;       // channel half for reductions / frag column
  const int b    = blockIdx.y;
  const int n0   = blockIdx.x * kT;
  const int n    = n0 + p;
  const int ca   = 4 * lane;        // channel group A: 0..127
  const int cb   = 128 + 4 * lane;  // channel group B: 128..255

  const float* fimg = imgF + (size_t)b * kC * kHW;
  const float* eimg = imgE + (size_t)b * kC * kHW;
  const float* pfb  = pf + (size_t)(b * kN + n0) * kC;

  // ---- Phase 1: stage rep (= points_feature tile) into LDS ----------------
#if USE_ASYNC_LDS
  for (int r = 0; r < kT; ++r) {
    const float* src = pfb + (size_t)r * kC;
    __builtin_amdgcn_global_load_async_to_lds_b128(
        (gv4i_p)(src + ca), (lv4i_p)&s_rep[r][ca], 0, 0);
    __builtin_amdgcn_global_load_async_to_lds_b128(
        (gv4i_p)(src + cb), (lv4i_p)&s_rep[r][cb], 0, 0);
  }
  wait_async0();
#else
  for (int r = 0; r < kT; ++r) {
    const float* src = pfb + (size_t)r * kC;
    *(float4*)&s_rep[r][ca] = ld4(src + ca);
    *(float4*)&s_rep[r][cb] = ld4(src + cb);
  }
#endif

  // ---- na = max(||rep_p||, eps) -------------------------------------------
  float na2 = 0.f;
  for (int i = 0; i < 128; i += 4) {
    float4 v = *(const float4*)&s_rep[p][half * 128 + i];
    na2 = fmaf(v.x, v.x, fmaf(v.y, v.y, fmaf(v.z, v.z, fmaf(v.w, v.w, na2))));
  }
  na2 += __shfl_xor(na2, 16, 32);
  const float na = fmaxf(sqrtf(na2), 1e-8f);

  // ---- Bilinear corners + weights for my point, all A offsets -------------
  const float px = pts[(size_t)(b * kN + n) * 2 + 0];
  const float py = pts[(size_t)(b * kN + n) * 2 + 1];

  int   q0[kA], q1[kA], q2[kA], q3[kA];
  float w0[kA], w1[kA], w2[kA], w3[kA];
  float sim[kA];

#pragma unroll
  for (int a = 0; a < kA; ++a) {
    const size_t ai = ((size_t)(b * kA + a) * kN + n) * 2;
    float gx = px + (ao[ai + 0] * 0.4f - 0.1f);
    float gy = py + (ao[ai + 1] * 0.4f - 0.1f);
    gx = fminf(fmaxf(gx, -1.f), 1.f);
    gy = fminf(fmaxf(gy, -1.f), 1.f);
    const float x  = (gx + 1.f) * (kW * 0.5f) - 0.5f;
    const float y  = (gy + 1.f) * (kH * 0.5f) - 0.5f;
    const float xf = floorf(x), yf = floorf(y);
    const float wx1 = x - xf, wy1 = y - yf;
    const float wx0 = 1.f - wx1, wy0 = 1.f - wy1;
    const int x0 = (int)xf, y0 = (int)yf, x1 = x0 + 1, y1 = y0 + 1;
    const float vx0 = (x0 >= 0 && x0 < kW) ? 1.f : 0.f;
    const float vx1 = (x1 >= 0 && x1 < kW) ? 1.f : 0.f;
    const float vy0 = (y0 >= 0 && y0 < kH) ? 1.f : 0.f;
    const float vy1 = (y1 >= 0 && y1 < kH) ? 1.f : 0.f;
    const int cx0 = clampi(x0, 0, kW - 1), cx1 = clampi(x1, 0, kW - 1);
    const int cy0 = clampi(y0, 0, kH - 1), cy1 = clampi(y1, 0, kH - 1);
    q0[a] = cy0 * kW + cx0;  w0[a] = wx0 * wy0 * vx0 * vy0;
    q1[a] = cy0 * kW + cx1;  w1[a] = wx1 * wy0 * vx1 * vy0;
    q2[a] = cy1 * kW + cx0;  w2[a] = wx0 * wy1 * vx0 * vy1;
    q3[a] = cy1 * kW + cx1;  w3[a] = wx1 * wy1 * vx1 * vy1;
  }

  // ---- Per offset: gather blended feat tile, WMMA dot/norm, similarity ----
#pragma unroll
  for (int a = 0; a < kA; ++a) {
    for (int pp = 0; pp < kT; ++pp) {   // coalesced wave-cooperative gather
      const int   g0 = __shfl(q0[a], pp, 32), g1 = __shfl(q1[a], pp, 32);
      const int   g2 = __shfl(q2[a], pp, 32), g3 = __shfl(q3[a], pp, 32);
      const float c0 = __shfl(w0[a], pp, 32), c1 = __shfl(w1[a], pp, 32);
      const float c2 = __shfl(w2[a], pp, 32), c3 = __shfl(w3[a], pp, 32);
      float4 acc0 = {0, 0, 0, 0}, acc1 = {0, 0, 0, 0};
      fma4(acc0, c0, ldpix<NHWC>(fimg, g0, ca)); fma4(acc1, c0, ldpix<NHWC>(fimg, g0, cb));
      fma4(acc0, c1, ldpix<NHWC>(fimg, g1, ca)); fma4(acc1, c1, ldpix<NHWC>(fimg, g1, cb));
      fma4(acc0, c2, ldpix<NHWC>(fimg, g2, ca)); fma4(acc1, c2, ldpix<NHWC>(fimg, g2, cb));
      fma4(acc0, c3, ldpix<NHWC>(fimg, g3, ca)); fma4(acc1, c3, ldpix<NHWC>(fimg, g3, cb));
      *(float4*)&s_feat[pp][ca] = acc0;
      *(float4*)&s_feat[pp][cb] = acc1;
    }

#if HAVE_WMMA_F32
    // A(16x4) and B(4x16) f32 fragments share the same lane layout:
    // lane L, reg v  <->  row L%16, k = kb + 2*(L/16) + v.
    v8f dacc = {0, 0, 0, 0, 0, 0, 0, 0};   // rep . feat^T
    v8f nacc = {0, 0, 0, 0, 0, 0, 0, 0};   // feat . feat^T
#pragma unroll 4
    for (int kb = 0; kb < kC; kb += 4) {
      const int col = kb + 2 * half;
      const v2f ra = *(const v2f*)&s_rep [p][col];
      const v2f fb = *(const v2f*)&s_feat[p][col];
      dacc = __builtin_amdgcn_wmma_f32_16x16x4_f32(false, ra, false, fb,
                                                   (short)0, dacc, false, false);
      nacc = __builtin_amdgcn_wmma_f32_16x16x4_f32(false, fb, false, fb,
                                                   (short)0, nacc, false, false);
    }
    // Branch-free diagonal extraction:
    //   diag p<8  lives at (lane=p,    reg p)
    //   diag p>=8 lives at (lane=p+16, reg p-8)
    // so lane L in {0..7, 24..31} owns diag element (L&7); every lane pulls
    // its point's diagonal with one wave32 shuffle.
    const int  r8   = lane & 7;
    const float dv  = dacc[r8];
    const float nv  = nacc[r8];
    const int  srcl = (p < 8) ? p : (p + 16);
    const float dot = __shfl(dv, srcl, 32);
    const float nb2 = __shfl(nv, srcl, 32);
#else
    float dp = 0.f, np2 = 0.f;
    for (int i = 0; i < 128; ++i) {
      const float rv = s_rep [p][half * 128 + i];
      const float fv = s_feat[p][half * 128 + i];
      dp = fmaf(rv, fv, dp); np2 = fmaf(fv, fv, np2);
    }
    dp  += __shfl_xor(dp, 16, 32);
    np2 += __shfl_xor(np2, 16, 32);
    const float dot = dp, nb2 = np2;
#endif
    const float nb = fmaxf(sqrtf(nb2), 1e-8f);
    sim[a] = dot / (na * nb);
  }

  // ---- Softmax over the A=5 offsets ---------------------------------------
  float m = sim[0];
#pragma unroll
  for (int a = 1; a < kA; ++a) m = fmaxf(m, sim[a]);
  float wgt[kA]; float wsum = 0.f;
#pragma unroll
  for (int a = 0; a < kA; ++a) { wgt[a] = __expf(sim[a] - m); wsum += wgt[a]; }
  const float winv = 1.f / wsum;
#pragma unroll
  for (int a = 0; a < kA; ++a) wgt[a] *= winv;

  // ---- Weighted pos-emb gather + residual, streamed straight to out ------
  for (int pp = 0; pp < kT; ++pp) {
    float4 acc0 = *(const float4*)&s_rep[pp][ca];   // residual = points_feature
    float4 acc1 = *(const float4*)&s_rep[pp][cb];
#pragma unroll
    for (int a = 0; a < kA; ++a) {
      const float ww = __shfl(wgt[a], pp, 32);
      const int   g0 = __shfl(q0[a], pp, 32), g1 = __shfl(q1[a], pp, 32);
      const int   g2 = __shfl(q2[a], pp, 32), g3 = __shfl(q3[a], pp, 32);
      const float c0 = ww * __shfl(w0[a], pp, 32), c1 = ww * __shfl(w1[a], pp, 32);
      const float c2 = ww * __shfl(w2[a], pp, 32), c3 = ww * __shfl(w3[a], pp, 32);
      fma4(acc0, c0, ldpix<NHWC>(eimg, g0, ca)); fma4(acc1, c0, ldpix<NHWC>(eimg, g0, cb));
      fma4(acc0, c1, ldpix<NHWC>(eimg, g1, ca)); fma4(acc1, c1, ldpix<NHWC>(eimg, g1, cb));
      fma4(acc0, c2, ldpix<NHWC>(eimg, g2, ca)); fma4(acc1, c2, ldpix<NHWC>(eimg, g2, cb));
      fma4(acc0, c3, ldpix<NHWC>(eimg, g3, ca)); fma4(acc1, c3, ldpix<NHWC>(eimg, g3, cb));
    }
    float* dst = out + (size_t)(b * kN + n0 + pp) * kC;
    *(float4*)(dst + ca) = acc0;
    *(float4*)(dst + cb) = acc1;
  }
}

} // anonymous namespace

extern "C" void kernel_launch(void* const* d_in, const int* in_sizes, int n_in,
                              void* d_out, int out_size, void* d_ws, size_t ws_size,
                              hipStream_t stream) {
  const float* x1  = (const float*)d_in[0];  // [B,C,H,W]
  const float* pf  = (const float*)d_in[1];  // [B,N,C]
  const float* pts = (const float*)d_in[2];  // [B,N,2]
  const float* pe  = (const float*)d_in[3];  // [B,C,H,W]
  const float* ao  = (const float*)d_in[4];  // [B,A,N,2]
  float* out = (float*)d_out;

  const size_t imgElems = (size_t)kB * kC * kHW;
  const bool nhwc = (d_ws != nullptr) && (ws_size >= 2 * imgElems * sizeof(float));

  if (nhwc) {
    float* wx = (float*)d_ws;
    float* we = wx + imgElems;
    dim3 tg(kHW / 32, kC / 32, kB), tb(32, 8);
    transpose_nchw_nhwc<<<tg, tb, 0, stream>>>(x1, wx);
    transpose_nchw_nhwc<<<tg, tb, 0, stream>>>(pe, we);
    soft_align_kernel<true><<<dim3(kN / kT, kB), 32, 0, stream>>>(wx, we, pf, pts, ao, out);
  } else {
    soft_align_kernel<false><<<dim3(kN / kT, kB), 32, 0, stream>>>(x1, pe, pf, pts, ao, out);
  }
  (void)in_sizes; (void)n_in; (void)out_size;
}